// Propagation_Attention_3367254360659
// MI455X (gfx1250) — compile-verified
//
#include <hip/hip_runtime.h>
#include <hip/hip_bf16.h>

#define BB 8
#define NN 1024
#define CC 256
#define HH 8
#define HD 32
#define TABLE 10

typedef float v2f __attribute__((ext_vector_type(2)));
typedef float v8f __attribute__((ext_vector_type(8)));

__device__ __forceinline__ v8f zero8() {
    v8f z = {0.f, 0.f, 0.f, 0.f, 0.f, 0.f, 0.f, 0.f};
    return z;
}

// D = A(16x4) * B(4x16) + C, all f32 (V_WMMA_F32_16X16X4_F32)
__device__ __forceinline__ v8f wmma4(v2f a, v2f b, v8f c) {
    return __builtin_amdgcn_wmma_f32_16x16x4_f32(false, a, false, b, (short)0, c,
                                                 false, false);
}

// Async global->LDS copy of 32 contiguous bytes (2 x B128).  INST_OFFSET is
// added to BOTH the LDS and global addresses (ISA 15.18.3), so one address
// pair covers both halves.  Tracked by ASYNCcnt.
__device__ __forceinline__ void async_copy32(void* lds_dst, const void* gsrc) {
    unsigned lds32 = (unsigned)(size_t)lds_dst;            // LDS offset (low 32)
    unsigned long long g = (unsigned long long)(size_t)gsrc;
    asm volatile(
        "global_load_async_to_lds_b128 %0, %1, off\n\t"
        "global_load_async_to_lds_b128 %0, %1, off offset:16"
        :
        : "v"(lds32), "v"(g)
        : "memory");
}

__device__ __forceinline__ void wait_async0() {
    asm volatile("s_wait_asynccnt 0x0" ::: "memory");
}

// ---------------------------------------------------------------------------
// Kernel 1: QKV projection.  X[8192,256] @ Wqkv[256,768] -> Q/K/V [B,H,N,HD]
// block = 128 threads (4 waves); block tile 64 rows x 64 cols; K staged 16-wide
// via GLOBAL_LOAD_ASYNC_TO_LDS_B128.
// ---------------------------------------------------------------------------
__global__ void qkv_gemm_kernel(const float* __restrict__ X,
                                const float* __restrict__ Wqkv,
                                float* __restrict__ Q, float* __restrict__ K,
                                float* __restrict__ V) {
    __shared__ float lds_x[64][16];
    __shared__ float lds_w[16][64];

    const int tid  = threadIdx.x;
    const int wv   = tid >> 5;
    const int lane = tid & 31;
    const int hf   = lane >> 4;   // which half of the wave
    const int col  = lane & 15;
    const int r0   = blockIdx.x * 64;
    const int c0   = blockIdx.y * 64;

    const int xrow = tid >> 1, xcs = (tid & 1) * 8;   // X tile 64x16
    const int wrow = tid >> 3, wcs = (tid & 7) * 8;   // W tile 16x64

    v8f acc[4];
#pragma unroll
    for (int t = 0; t < 4; ++t) acc[t] = zero8();

    for (int k0 = 0; k0 < CC; k0 += 16) {
        __syncthreads();   // previous iteration's readers done
        async_copy32(&lds_x[xrow][xcs],
                     X + (size_t)(r0 + xrow) * CC + k0 + xcs);
        async_copy32(&lds_w[wrow][wcs],
                     Wqkv + (size_t)(k0 + wrow) * (3 * CC) + c0 + wcs);
        wait_async0();
        __syncthreads();
#pragma unroll
        for (int kk = 0; kk < 4; ++kk) {
            v2f a;
            a.x = lds_x[wv * 16 + col][kk * 4 + hf * 2];
            a.y = lds_x[wv * 16 + col][kk * 4 + hf * 2 + 1];
#pragma unroll
            for (int t = 0; t < 4; ++t) {
                v2f b;
                b.x = lds_w[kk * 4 + hf * 2][t * 16 + col];
                b.y = lds_w[kk * 4 + hf * 2 + 1][t * 16 + col];
                acc[t] = wmma4(a, b, acc[t]);
            }
        }
    }

    // scatter into Q/K/V [B,H,N,HD]
#pragma unroll
    for (int t = 0; t < 4; ++t) {
        int gcol  = c0 + t * 16 + col;
        int which = gcol >> 8;      // 0=q 1=k 2=v
        int rem   = gcol & 255;
        int h     = rem >> 5;
        int d     = rem & 31;
        float* dst = (which == 0) ? Q : ((which == 1) ? K : V);
#pragma unroll
        for (int r = 0; r < 8; ++r) {
            int grow = r0 + wv * 16 + hf * 8 + r;
            int b    = grow >> 10;
            int n    = grow & (NN - 1);
            dst[(((size_t)(b * HH + h)) * NN + n) * HD + d] = acc[t][r];
        }
    }
}

// ---------------------------------------------------------------------------
// Kernel 2: flash attention per (b,h).  Block = 4 waves; each wave owns 16
// query rows; streams 64-key blocks with online softmax; WMMA for S and P*V.
// ---------------------------------------------------------------------------
__global__ void attn_kernel(const float* __restrict__ Q,
                            const float* __restrict__ K,
                            const float* __restrict__ V,
                            const int* __restrict__ RP,
                            const int* __restrict__ rel_len,
                            const float* __restrict__ bias_table,
                            float* __restrict__ Xb) {
    __shared__ float lds_q[4][16][HD];   // pre-scaled Q, per wave
    __shared__ float lds_p[4][16][64];   // P tile round-trip, per wave
    __shared__ float lut[TABLE];

    const int tid  = threadIdx.x;
    const int wv   = tid >> 5;
    const int lane = tid & 31;
    const int hf   = lane >> 4;
    const int col  = lane & 15;

    const int nqb = NN / 64;
    const int bh  = blockIdx.x / nqb;
    const int qb  = blockIdx.x % nqb;
    const int b   = bh / HH;
    const int h   = bh % HH;
    const int q0  = qb * 64 + wv * 16;   // first query row of this wave

    if (tid < TABLE) {
        int ml = (int)((float)rel_len[b] * 0.5f);
        lut[tid] = bias_table[tid * HH + h] + ((tid > ml) ? -100.0f : 0.0f);
    }

    const float scale = 0.17677669529663687f;  // 1/sqrt(32)
    {
        const float* qsrc = Q + ((size_t)bh * NN + q0) * HD;
        for (int i = lane; i < 16 * HD; i += 32)
            lds_q[wv][i >> 5][i & 31] = qsrc[i] * scale;
    }
    __syncthreads();

    float m_run[8], l_run[8];
#pragma unroll
    for (int r = 0; r < 8; ++r) { m_run[r] = -3.0e38f; l_run[r] = 0.f; }
    v8f o[2];
    o[0] = zero8();
    o[1] = zero8();

    const float* Kbase = K + (size_t)bh * NN * HD;
    const float* Vbase = V + (size_t)bh * NN * HD;
    const int*   RPb   = RP + (size_t)b * NN * NN;

    for (int j0 = 0; j0 < NN; j0 += 64) {
        // prefetch next key block's streams (global_prefetch_b8)
        if (j0 + 64 < NN) {
            __builtin_prefetch(Kbase + (size_t)(j0 + 64 + (lane << 1)) * HD, 0, 0);
            __builtin_prefetch(Vbase + (size_t)(j0 + 64 + (lane << 1)) * HD, 0, 0);
            __builtin_prefetch(
                RPb + (size_t)(q0 + hf * 8) * NN + (j0 + 64) + col * 4, 0, 0);
        }
        // ---- S tiles: Qs[16x32] @ K^T -> 4 tiles of 16x16 ----
        v8f s[4];
#pragma unroll
        for (int t = 0; t < 4; ++t) {
            v8f a8 = zero8();
            const float* kp = Kbase + (size_t)(j0 + t * 16 + col) * HD;
#pragma unroll
            for (int kk = 0; kk < 8; ++kk) {
                v2f af, bf;
                af.x = lds_q[wv][col][kk * 4 + hf * 2];
                af.y = lds_q[wv][col][kk * 4 + hf * 2 + 1];
                bf.x = kp[kk * 4 + hf * 2];
                bf.y = kp[kk * 4 + hf * 2 + 1];
                a8 = wmma4(af, bf, a8);
            }
            s[t] = a8;
        }
        // ---- add positional/mask bias (32 MB tensor, read once per head) ----
#pragma unroll
        for (int t = 0; t < 4; ++t) {
#pragma unroll
            for (int r = 0; r < 8; ++r) {
                int m  = hf * 8 + r;
                int rp = RPb[(size_t)(q0 + m) * NN + (j0 + t * 16 + col)];
                s[t][r] += lut[rp];
            }
        }
        // ---- online softmax: per-row max via 16-lane butterflies ----
#pragma unroll
        for (int r = 0; r < 8; ++r) {
            float v = fmaxf(fmaxf(s[0][r], s[1][r]), fmaxf(s[2][r], s[3][r]));
            v = fmaxf(v, __shfl_xor(v, 1, 16));
            v = fmaxf(v, __shfl_xor(v, 2, 16));
            v = fmaxf(v, __shfl_xor(v, 4, 16));
            v = fmaxf(v, __shfl_xor(v, 8, 16));
            float m_new  = fmaxf(m_run[r], v);
            float alpha  = __expf(m_run[r] - m_new);
            float rs = 0.f;
#pragma unroll
            for (int t = 0; t < 4; ++t) {
                float p = __expf(s[t][r] - m_new);
                s[t][r] = p;
                rs += p;
            }
            rs += __shfl_xor(rs, 1, 16);
            rs += __shfl_xor(rs, 2, 16);
            rs += __shfl_xor(rs, 4, 16);
            rs += __shfl_xor(rs, 8, 16);
            l_run[r] = l_run[r] * alpha + rs;
            m_run[r] = m_new;
            o[0][r] *= alpha;   // rescale accumulator: same (hf,r) row mapping
            o[1][r] *= alpha;
        }
        // ---- P -> LDS (C-layout to A-layout round trip, wave-private) ----
#pragma unroll
        for (int t = 0; t < 4; ++t)
#pragma unroll
            for (int r = 0; r < 8; ++r)
                lds_p[wv][hf * 8 + r][t * 16 + col] = s[t][r];
        // ---- O += P[16x64] @ V[64x32] ----
#pragma unroll
        for (int dt = 0; dt < 2; ++dt) {
            v8f a8 = o[dt];
#pragma unroll
            for (int t = 0; t < 4; ++t) {
#pragma unroll
                for (int kk = 0; kk < 4; ++kk) {
                    v2f af, bf;
                    af.x = lds_p[wv][col][t * 16 + kk * 4 + hf * 2];
                    af.y = lds_p[wv][col][t * 16 + kk * 4 + hf * 2 + 1];
                    const float* vp =
                        Vbase + (size_t)(j0 + t * 16 + kk * 4 + hf * 2) * HD +
                        dt * 16 + col;
                    bf.x = vp[0];
                    bf.y = vp[HD];
                    a8 = wmma4(af, bf, a8);
                }
            }
            o[dt] = a8;
        }
    }

    // ---- normalize and write to xbuf [B,N,C] with c = h*32 + d ----
#pragma unroll
    for (int r = 0; r < 8; ++r) {
        float inv = 1.0f / l_run[r];
        int m = hf * 8 + r;
        size_t row = (size_t)b * NN + (q0 + m);
        Xb[row * CC + h * HD + col]      = o[0][r] * inv;
        Xb[row * CC + h * HD + 16 + col] = o[1][r] * inv;
    }
}

// ---------------------------------------------------------------------------
// Kernel 3: output projection.  Xb[8192,256] @ Wp[256,256] + bias -> Out
// ---------------------------------------------------------------------------
__global__ void proj_gemm_kernel(const float* __restrict__ Xb,
                                 const float* __restrict__ Wp,
                                 const float* __restrict__ bp,
                                 float* __restrict__ Out) {
    __shared__ float lds_x[64][16];
    __shared__ float lds_w[16][64];

    const int tid  = threadIdx.x;
    const int wv   = tid >> 5;
    const int lane = tid & 31;
    const int hf   = lane >> 4;
    const int col  = lane & 15;
    const int r0   = blockIdx.x * 64;
    const int c0   = blockIdx.y * 64;

    const int xrow = tid >> 1, xcs = (tid & 1) * 8;
    const int wrow = tid >> 3, wcs = (tid & 7) * 8;

    v8f acc[4];
#pragma unroll
    for (int t = 0; t < 4; ++t) acc[t] = zero8();

    for (int k0 = 0; k0 < CC; k0 += 16) {
        __syncthreads();
        async_copy32(&lds_x[xrow][xcs],
                     Xb + (size_t)(r0 + xrow) * CC + k0 + xcs);
        async_copy32(&lds_w[wrow][wcs],
                     Wp + (size_t)(k0 + wrow) * CC + c0 + wcs);
        wait_async0();
        __syncthreads();
#pragma unroll
        for (int kk = 0; kk < 4; ++kk) {
            v2f a;
            a.x = lds_x[wv * 16 + col][kk * 4 + hf * 2];
            a.y = lds_x[wv * 16 + col][kk * 4 + hf * 2 + 1];
#pragma unroll
            for (int t = 0; t < 4; ++t) {
                v2f b;
                b.x = lds_w[kk * 4 + hf * 2][t * 16 + col];
                b.y = lds_w[kk * 4 + hf * 2 + 1][t * 16 + col];
                acc[t] = wmma4(a, b, acc[t]);
            }
        }
    }

#pragma unroll
    for (int t = 0; t < 4; ++t) {
        int gcol = c0 + t * 16 + col;
        float bias = bp[gcol];
#pragma unroll
        for (int r = 0; r < 8; ++r) {
            int grow = r0 + wv * 16 + hf * 8 + r;
            Out[(size_t)grow * CC + gcol] = acc[t][r] + bias;
        }
    }
}

// ---------------------------------------------------------------------------
extern "C" void kernel_launch(void* const* d_in, const int* in_sizes, int n_in,
                              void* d_out, int out_size, void* d_ws,
                              size_t ws_size, hipStream_t stream) {
    const float* X          = (const float*)d_in[0];  // [8,1024,256]
    const int*   RP         = (const int*)d_in[1];    // [8,1024,1024]
    const int*   rel_len    = (const int*)d_in[2];    // [8]
    const float* Wqkv       = (const float*)d_in[3];  // [256,768]
    const float* Wproj      = (const float*)d_in[4];  // [256,256]
    const float* bproj      = (const float*)d_in[5];  // [256]
    const float* bias_table = (const float*)d_in[6];  // [10,8]
    float*       Out        = (float*)d_out;          // [8,1024,256]

    const size_t headElems = (size_t)BB * HH * NN * HD;  // 2M floats
    float* Qb = (float*)d_ws;
    float* Kb = Qb + headElems;
    float* Vb = Kb + headElems;
    float* Xb = Vb + headElems;  // [8,1024,256] attention output, pre-proj

    dim3 g1((BB * NN) / 64, (3 * CC) / 64);  // 128 x 12
    qkv_gemm_kernel<<<g1, 128, 0, stream>>>(X, Wqkv, Qb, Kb, Vb);

    dim3 g2(BB * HH * (NN / 64));            // 1024 blocks
    attn_kernel<<<g2, 128, 0, stream>>>(Qb, Kb, Vb, RP, rel_len, bias_table, Xb);

    dim3 g3((BB * NN) / 64, CC / 64);        // 128 x 4
    proj_gemm_kernel<<<g3, 128, 0, stream>>>(Xb, Wproj, bproj, Out);
}